// GANConv_25357486916125
// MI455X (gfx1250) — compile-verified
//
#include <hip/hip_runtime.h>

typedef __attribute__((ext_vector_type(2))) float v2f;
typedef __attribute__((ext_vector_type(8))) float v8f;

#define N_NODES 100000
#define N_EDGES 1600000
#define D_FEAT  64
#define D_OUT   256

// ---------------------------------------------------------------------------
// Kernel 1: zero the aggregation scratch (float4 stores, 1.6M threads)
// ---------------------------------------------------------------------------
__global__ void zero_agg(float4* __restrict__ agg4, int n4) {
  int i = blockIdx.x * blockDim.x + threadIdx.x;
  if (i < n4) agg4[i] = make_float4(0.f, 0.f, 0.f, 0.f);
}

// ---------------------------------------------------------------------------
// Kernel 2: edge scatter-add. 16 threads per edge, float4 per thread.
// agg[row[e]][:] += x[col[e]][:]   (L2-resident atomics)
// ---------------------------------------------------------------------------
__global__ void scatter_add(const float* __restrict__ x,
                            const int*   __restrict__ ei,
                            float*       __restrict__ agg) {
  long long t = (long long)blockIdx.x * blockDim.x + threadIdx.x;
  int e = (int)(t >> 4);
  if (e >= N_EDGES) return;
  int f = ((int)t & 15) * 4;
  int r = ei[e];            // destination (segment id)
  int c = ei[N_EDGES + e];  // source node
  const float4 v = *(const float4*)(x + (size_t)c * D_FEAT + f);
  float* dst = agg + (size_t)r * D_FEAT + f;
  atomicAdd(dst + 0, v.x);
  atomicAdd(dst + 1, v.y);
  atomicAdd(dst + 2, v.z);
  atomicAdd(dst + 3, v.w);
}

// ---------------------------------------------------------------------------
// Kernel 3: fused residual + GEMM + bias using V_WMMA_F32_16X16X4_F32.
// Block = 256 threads = 8 wave32. Each block: 16 output rows.
// Each wave: one 16-wide column tile. grid = (6250, 2) -> 100000 x 256.
// ---------------------------------------------------------------------------
__global__ __launch_bounds__(256) void fused_gemm(
    const float* __restrict__ x,
    const float* __restrict__ agg,
    const float* __restrict__ W,   // [64, 256] row-major (in, out)
    const float* __restrict__ b,   // [256]
    float*       __restrict__ out) // [100000, 256]
{
  // 16 rows x 64 K of (x + agg), padded row stride 65 to avoid bank conflicts
  __shared__ float Alds[16 * 65];

  const int row0 = blockIdx.x * 16;

  // Stage residual tile into LDS
  for (int idx = threadIdx.x; idx < 16 * 64; idx += 256) {
    int m = idx >> 6;
    int k = idx & 63;
    size_t g = (size_t)(row0 + m) * D_FEAT + k;
    Alds[m * 65 + k] = x[g] + agg[g];
  }
  __syncthreads();

  const int lane = threadIdx.x & 31;  // wave32
  const int wave = threadIdx.x >> 5;  // 0..7
  const int m16  = lane & 15;
  const int half = lane >> 4;         // 0 or 1
  const int n    = blockIdx.y * 128 + wave * 16 + m16;

  v8f c = {0.f, 0.f, 0.f, 0.f, 0.f, 0.f, 0.f, 0.f};

#pragma unroll
  for (int kk = 0; kk < D_FEAT; kk += 4) {
    const int k0 = kk + 2 * half;     // lane's K pair base
    v2f a, bb;
    // A 16x4 f32: lane m = lane%16, VGPR v holds K = 2*(lane/16)+v
    a.x = Alds[m16 * 65 + k0];
    a.y = Alds[m16 * 65 + k0 + 1];
    // B 4x16 f32: lane n = lane%16, VGPR v holds K = 2*(lane/16)+v
    bb.x = W[(size_t)(k0)     * D_OUT + n];
    bb.y = W[(size_t)(k0 + 1) * D_OUT + n];
    c = __builtin_amdgcn_wmma_f32_16x16x4_f32(
        /*neg_a=*/false, a, /*neg_b=*/false, bb,
        /*c_mod=*/(short)0, c, /*reuse_a=*/false, /*reuse_b=*/false);
  }

  const float bias = b[n];
#pragma unroll
  for (int r = 0; r < 8; ++r) {
    int m = r + 8 * half;             // C/D layout: VGPR r, halves of wave
    out[(size_t)(row0 + m) * D_OUT + n] = c[r] + bias;
  }
}

// ---------------------------------------------------------------------------
extern "C" void kernel_launch(void* const* d_in, const int* in_sizes, int n_in,
                              void* d_out, int out_size, void* d_ws, size_t ws_size,
                              hipStream_t stream) {
  const float* x  = (const float*)d_in[0];   // [100000, 64]
  const int*   ei = (const int*)d_in[1];     // [2, 1600000]
  const float* W  = (const float*)d_in[2];   // [64, 256]
  const float* b  = (const float*)d_in[3];   // [256]
  float* out = (float*)d_out;                // [100000, 256]
  float* agg = (float*)d_ws;                 // 100000*64 floats = 25.6 MB

  // 1) zero agg
  {
    int n4 = (N_NODES * D_FEAT) / 4;  // 1.6M float4
    int blocks = (n4 + 255) / 256;
    zero_agg<<<blocks, 256, 0, stream>>>((float4*)agg, n4);
  }
  // 2) scatter-add over edges
  {
    long long threads = (long long)N_EDGES * 16;
    int blocks = (int)((threads + 255) / 256);
    scatter_add<<<blocks, 256, 0, stream>>>(x, ei, agg);
  }
  // 3) fused residual + GEMM + bias
  {
    dim3 grid(N_NODES / 16, D_OUT / 128, 1);  // 6250 x 2
    fused_gemm<<<grid, 256, 0, stream>>>(x, agg, W, b, out);
  }
}